// Nms3d_63737314673340
// MI455X (gfx1250) — compile-verified
//
#include <hip/hip_runtime.h>
#include <stdint.h>

#define NDET 4096
#define WORDS (NDET / 32)   // 128 mask words per row
#define IOU_TRF 0.3f
#define SCORE_TRF 0.1f
#define EPSF 1e-9f

typedef __attribute__((ext_vector_type(2))) float v2f;
typedef __attribute__((ext_vector_type(8))) float v8f;

// ---------------- K1: in-LDS bitonic sort, key = (score_bits, ~idx), descending ----------------
__global__ __launch_bounds__(512) void nms_sort(const float* __restrict__ results,
                                                uint32_t* __restrict__ order) {
    __shared__ uint64_t keys[NDET];
    const int tid = threadIdx.x;
    for (int i = tid; i < NDET; i += 512) {
        uint32_t sb = __float_as_uint(results[(size_t)i * 8]);   // scores in [0,1): bits monotonic
        keys[i] = ((uint64_t)sb << 32) | (uint64_t)(0xFFFFFFFFu - (uint32_t)i);
    }
    __syncthreads();
    for (int k = 2; k <= NDET; k <<= 1) {
        for (int j = k >> 1; j > 0; j >>= 1) {
            for (int i = tid; i < NDET; i += 512) {
                int ixj = i ^ j;
                if (ixj > i) {
                    uint64_t a = keys[i], b = keys[ixj];
                    bool desc = ((i & k) == 0);
                    bool sw = desc ? (a < b) : (a > b);
                    if (sw) { keys[i] = b; keys[ixj] = a; }
                }
            }
            __syncthreads();
        }
    }
    for (int i = tid; i < NDET; i += 512)
        order[i] = 0xFFFFFFFFu - (uint32_t)(keys[i] & 0xFFFFFFFFull);
}

// ---------------- K2: gather sorted boxes -> SoA lo/hi/vol/score ----------------
__global__ __launch_bounds__(256) void nms_prep(const float* __restrict__ results,
                                                const uint32_t* __restrict__ order,
                                                float* __restrict__ sscore,
                                                float* __restrict__ lox, float* __restrict__ loy,
                                                float* __restrict__ loz, float* __restrict__ hix,
                                                float* __restrict__ hiy, float* __restrict__ hiz,
                                                float* __restrict__ vol) {
    int i = blockIdx.x * blockDim.x + threadIdx.x;
    if (i >= NDET) return;
    const float* r = results + (size_t)order[i] * 8;
    float s = r[0];
    float cx = r[2], cy = r[3], cz = r[4];
    float dx = r[5], dy = r[6], dz = r[7];
    sscore[i] = s;
    lox[i] = cx - 0.5f * dx;  hix[i] = cx + 0.5f * dx;
    loy[i] = cy - 0.5f * dy;  hiy[i] = cy + 0.5f * dy;
    loz[i] = cz - 0.5f * dz;  hiz[i] = cz + 0.5f * dz;
    vol[i] = dx * dy * dz;
}

// ---------------- K3: suppression bitmask; WMMA computes union outer-sum per 16x16 tile ----------
// One wave handles 16 rows x 32 cols (two 16x16 WMMA tiles -> one 32-bit mask word per row).
// Predicate uses division-free form: inter > IOU_TR*(union+eps); union >= max(vol) >= 1 > 0
// (dims in [1,9]) so this is equivalent to inter/(union+eps) > IOU_TR without the IEEE div chain.
__global__ __launch_bounds__(256) void nms_mask(const float* __restrict__ lox, const float* __restrict__ loy,
                                                const float* __restrict__ loz, const float* __restrict__ hix,
                                                const float* __restrict__ hiy, const float* __restrict__ hiz,
                                                const float* __restrict__ vol,
                                                uint32_t* __restrict__ mask) {
    const int lane = threadIdx.x & 31;
    const int wave = blockIdx.x * (blockDim.x >> 5) + (threadIdx.x >> 5);
    const int rowTile = wave >> 7;       // 0..255
    const int colw    = wave & 127;      // mask word index, 0..127
    const int i0 = rowTile * 16;
    const int j0 = colw * 32;
    const int hf = lane >> 4;            // half-wave id
    const int m  = lane & 15;

    // A row m = [vol_r, 1, 1, 0]; B col n = [1, vol_c, 0, 0] with zero padding placed so that
    // D[m][n] = vol_r[m] + vol_c[n] under either plausible 32-bit A/B K-striping.
    float vr  = vol[i0 + m];
    float vc0 = vol[j0 + m];
    float vc1 = vol[j0 + 16 + m];
    v2f a;  a.x  = hf ? 1.0f : vr;   a.y  = hf ? 0.0f : 1.0f;
    v2f b0; b0.x = hf ? 0.0f : 1.0f; b0.y = hf ? 0.0f : vc0;
    v2f b1; b1.x = hf ? 0.0f : 1.0f; b1.y = hf ? 0.0f : vc1;
    v8f z = {};
    // D[v][lane] = vol_row[v + 8*hf] + vol_col[j0(+16) + m]
    v8f u0 = __builtin_amdgcn_wmma_f32_16x16x4_f32(false, a, false, b0, (short)0, z, false, false);
    v8f u1 = __builtin_amdgcn_wmma_f32_16x16x4_f32(false, a, false, b1, (short)0, z, false, false);

    // This lane's two columns
    const int c0 = j0 + m, c1 = j0 + 16 + m;
    float c0lx = lox[c0], c0ly = loy[c0], c0lz = loz[c0];
    float c0hx = hix[c0], c0hy = hiy[c0], c0hz = hiz[c0];
    float c1lx = lox[c1], c1ly = loy[c1], c1lz = loz[c1];
    float c1hx = hix[c1], c1hy = hiy[c1], c1hz = hiz[c1];

#pragma unroll
    for (int v = 0; v < 8; ++v) {
        const int R = i0 + v + 8 * hf;   // row this lane's D element belongs to
        float rlx = lox[R], rly = loy[R], rlz = loz[R];
        float rhx = hix[R], rhy = hiy[R], rhz = hiz[R];

        float ix0 = fmaxf(fminf(rhx, c0hx) - fmaxf(rlx, c0lx), 0.0f);
        float iy0 = fmaxf(fminf(rhy, c0hy) - fmaxf(rly, c0ly), 0.0f);
        float iz0 = fmaxf(fminf(rhz, c0hz) - fmaxf(rlz, c0lz), 0.0f);
        float in0 = ix0 * iy0 * iz0;
        float un0 = u0[v] - in0;                       // vol_r + vol_c - inter (WMMA outer-sum)
        bool  p0  = in0 > IOU_TRF * (un0 + EPSF);      // division-free IoU threshold

        float ix1 = fmaxf(fminf(rhx, c1hx) - fmaxf(rlx, c1lx), 0.0f);
        float iy1 = fmaxf(fminf(rhy, c1hy) - fmaxf(rly, c1ly), 0.0f);
        float iz1 = fmaxf(fminf(rhz, c1hz) - fmaxf(rlz, c1lz), 0.0f);
        float in1 = ix1 * iy1 * iz1;
        float un1 = u1[v] - in1;
        bool  p1  = in1 > IOU_TRF * (un1 + EPSF);

        uint32_t bal0 = __builtin_amdgcn_ballot_w32(p0);  // [15:0]=row v, [31:16]=row v+8 (cols j0..+15)
        uint32_t bal1 = __builtin_amdgcn_ballot_w32(p1);  // same rows, cols j0+16..+31
        if (lane == 0) {
            uint32_t wlo = (bal0 & 0xFFFFu) | (bal1 << 16);
            uint32_t whi = (bal0 >> 16) | (bal1 & 0xFFFF0000u);
            mask[(size_t)(i0 + v) * WORDS + colw]     = wlo;
            mask[(size_t)(i0 + v + 8) * WORDS + colw] = whi;
        }
    }
}

// ---------------- K4: serial greedy scan over bitmask (one workgroup) ----------------
__global__ __launch_bounds__(128) void nms_scan(const float* __restrict__ sscore,
                                                const uint32_t* __restrict__ mask,
                                                uint32_t* __restrict__ keepw) {
    __shared__ uint32_t s_active[WORDS];
    __shared__ uint32_t s_keep[WORDS];
    const int t = threadIdx.x;           // 0..127, one mask word each
    uint32_t w = 0;
#pragma unroll
    for (int b = 0; b < 32; ++b)
        if (sscore[t * 32 + b] >= SCORE_TRF) w |= (1u << b);
    s_active[t] = w;
    s_keep[t]   = 0;
    __syncthreads();

    for (int i = 0; i < NDET; ++i) {
        const int wi = i >> 5, bi = i & 31;
        uint32_t cur = (s_active[wi] >> bi) & 1u;      // uniform read
        __syncthreads();
        if (cur) {
            s_active[t] &= ~mask[(size_t)i * WORDS + t];  // bits at j<=i are already decided: harmless
            if (t == 0) s_keep[wi] |= (1u << bi);
        }
        __syncthreads();
    }
    keepw[t] = s_keep[t];
}

// ---------------- K5: emit kept sorted rows, zeros elsewhere ----------------
__global__ __launch_bounds__(256) void nms_out(const float* __restrict__ results,
                                               const uint32_t* __restrict__ order,
                                               const uint32_t* __restrict__ keepw,
                                               float* __restrict__ out) {
    int i = blockIdx.x * blockDim.x + threadIdx.x;
    if (i >= NDET) return;
    bool k = (keepw[i >> 5] >> (i & 31)) & 1u;
    const float4* src = (const float4*)(results + (size_t)order[i] * 8);
    float4 a = src[0], b = src[1];
    float4 zz = make_float4(0.f, 0.f, 0.f, 0.f);
    float4* dst = (float4*)(out + (size_t)i * 8);
    dst[0] = k ? a : zz;
    dst[1] = k ? b : zz;
}

extern "C" void kernel_launch(void* const* d_in, const int* in_sizes, int n_in,
                              void* d_out, int out_size, void* d_ws, size_t ws_size,
                              hipStream_t stream) {
    const float* results = (const float*)d_in[0];
    char* ws = (char*)d_ws;
    // workspace layout (all fully written before read)
    uint32_t* order  = (uint32_t*)(ws + 0);            // 16 KB
    float*    sscore = (float*)(ws + 16 * 1024);       // 16 KB
    float*    lox    = (float*)(ws + 32 * 1024);
    float*    loy    = (float*)(ws + 48 * 1024);
    float*    loz    = (float*)(ws + 64 * 1024);
    float*    hix    = (float*)(ws + 80 * 1024);
    float*    hiy    = (float*)(ws + 96 * 1024);
    float*    hiz    = (float*)(ws + 112 * 1024);
    float*    vol    = (float*)(ws + 128 * 1024);
    uint32_t* keepw  = (uint32_t*)(ws + 144 * 1024);   // 512 B
    uint32_t* mask   = (uint32_t*)(ws + 160 * 1024);   // 4096*128*4 = 2 MB
    float*    out    = (float*)d_out;

    nms_sort<<<1, 512, 0, stream>>>(results, order);
    nms_prep<<<NDET / 256, 256, 0, stream>>>(results, order, sscore, lox, loy, loz, hix, hiy, hiz, vol);
    // 32768 wave-tiles (16 rows x 32 cols each), 8 waves per 256-thread block
    nms_mask<<<4096, 256, 0, stream>>>(lox, loy, loz, hix, hiy, hiz, vol, mask);
    nms_scan<<<1, 128, 0, stream>>>(sscore, mask, keepw);
    nms_out<<<NDET / 256, 256, 0, stream>>>(results, order, keepw, out);
}